// GraphEncoder_72919954751600
// MI455X (gfx1250) — compile-verified
//
#include <hip/hip_runtime.h>
#include <hip/hip_bf16.h>
#include <math.h>

// ---------------------------------------------------------------------------
// GraphEncoder for MI455X (gfx1250, wave32).
// Dense 128x128 layers via V_WMMA_F32_16X16X4_F32 with the A-tile staged in
// LDS through GLOBAL_LOAD_ASYNC_TO_LDS_B128 (ASYNCcnt DMA). GATv2 edge
// softmax via global float atomics, PMA only on final layer.
// ---------------------------------------------------------------------------

#define BB     8
#define NN     2048
#define EE     32768
#define NP1    2049
#define ROWS   (BB * NP1)      // 16392
#define MP     16400           // padded to multiple of 16
#define DD     128
#define HH     8
#define DH     16
#define APITCH 132             // LDS row pitch (floats): 16B aligned, bank-safe

typedef __attribute__((ext_vector_type(2))) float v2f;
typedef __attribute__((ext_vector_type(8))) float v8f;

// ---------- helpers ----------
__device__ __forceinline__ float wave_red_add(float v) {
  #pragma unroll
  for (int m = 16; m >= 1; m >>= 1) v += __shfl_xor(v, m, 32);
  return v;
}

__device__ __forceinline__ float mish_f(float x) {
  float sp = (x > 20.0f) ? x : log1pf(__expf(x));
  return x * tanhf(sp);
}

// monotonic float<->uint encoding for atomic max with negatives
__device__ __forceinline__ unsigned f2ord(float f) {
  unsigned u = __float_as_uint(f);
  return (u & 0x80000000u) ? ~u : (u | 0x80000000u);
}
__device__ __forceinline__ float ord2f(unsigned u) {
  unsigned v = (u & 0x80000000u) ? (u & 0x7FFFFFFFu) : ~u;
  return __uint_as_float(v);
}
#define ORD_NEG_INF 0x007FFFFFu   // f2ord(-inf)

// ---------- fills ----------
__global__ void fill_f32_kernel(float* p, float v, size_t n) {
  size_t i = (size_t)blockIdx.x * blockDim.x + threadIdx.x;
  size_t st = (size_t)gridDim.x * blockDim.x;
  for (; i < n; i += st) p[i] = v;
}
__global__ void fill_u32_kernel(unsigned* p, unsigned v, size_t n) {
  size_t i = (size_t)blockIdx.x * blockDim.x + threadIdx.x;
  size_t st = (size_t)gridDim.x * blockDim.x;
  for (; i < n; i += st) p[i] = v;
}

// ---------- embed1: (B,Np1,16) -> dense(16->128) -> rmsnorm -> mish ----------
__global__ __launch_bounds__(256) void embed1_kernel(
    const float* __restrict__ nf, const float* __restrict__ W,
    const float* __restrict__ bvec, const float* __restrict__ norm1,
    float* __restrict__ out) {
  int wid = (int)((blockIdx.x * blockDim.x + threadIdx.x) >> 5);
  if (wid >= ROWS) return;
  int lane = threadIdx.x & 31;
  int b = wid / NP1, n = wid - b * NP1;
  float x[16];
  if (n < NN) {
    const float* xp = nf + ((size_t)b * NN + n) * 16;
    #pragma unroll
    for (int i = 0; i < 16; ++i) x[i] = xp[i];
  } else {
    #pragma unroll
    for (int i = 0; i < 16; ++i) x[i] = 0.0f;   // pad node
  }
  int j0 = lane * 4;
  float y[4];
  #pragma unroll
  for (int c = 0; c < 4; ++c) {
    int j = j0 + c;
    float acc = bvec[j];
    #pragma unroll
    for (int i = 0; i < 16; ++i) acc += x[i] * W[i * DD + j];
    y[c] = acc;
  }
  float ss = y[0]*y[0] + y[1]*y[1] + y[2]*y[2] + y[3]*y[3];
  ss = wave_red_add(ss);
  float r = rsqrtf(ss * (1.0f / DD) + 1e-6f);
  float* op = out + (size_t)wid * DD + j0;
  #pragma unroll
  for (int c = 0; c < 4; ++c) op[c] = mish_f(y[c] * r * norm1[j0 + c]);
}

// ---------- GEMM: C = A(MPx128) @ W(128x128) + bias, fp32 WMMA -------------
// A tile (16x128) staged once per workgroup into LDS via async DMA; all 8
// waves (one 16-col tile each) consume it from LDS -> 8x less A traffic.
__global__ __launch_bounds__(256) void gemm128_kernel(
    const float* __restrict__ A, const float* __restrict__ W,
    const float* __restrict__ bias, float* __restrict__ C) {
  __shared__ float As[16][APITCH];
  int wave = threadIdx.x >> 5;          // 0..7 -> N tile
  int lane = threadIdx.x & 31;
  int row0 = blockIdx.x * 16;
  int col0 = wave * 16;
  int lrow  = lane & 15;
  int khalf = (lane >> 4) << 1;         // 0 or 2

  // ---- async-stage A tile: 512 float4 slots, 2 per thread ----
  {
    int slot = threadIdx.x;             // 0..255, then +256 -> full coverage
    #pragma unroll
    for (int it = 0; it < 2; ++it, slot += 256) {
      int arow = slot >> 5;             // 32 float4 per row
      int ac4  = slot & 31;
      const float* gp = A + (size_t)(row0 + arow) * DD + ac4 * 4;
      // ptrtoint of the real LDS address: low 32 bits == wave-relative LDS
      // offset, and it escapes As so the "memory"-clobbering asm is known to
      // write it (keeps the allocation + the ds reads alive).
      unsigned lds_off = (unsigned)(size_t)&As[arow][ac4 * 4];
      asm volatile("global_load_async_to_lds_b128 %0, %1, off"
                   :: "v"(lds_off), "v"(gp) : "memory");
    }
    asm volatile("s_wait_asynccnt 0" ::: "memory");
    __syncthreads();
  }

  v8f acc = {};
  #pragma unroll 4
  for (int kk = 0; kk < DD; kk += 4) {
    v2f a, b;
    float2 av = *(const float2*)&As[lrow][kk + khalf];  // ds_load_b64
    a.x = av.x;
    a.y = av.y;
    b.x = W[(size_t)(kk + khalf)     * DD + col0 + lrow];
    b.y = W[(size_t)(kk + khalf + 1) * DD + col0 + lrow];
    acc = __builtin_amdgcn_wmma_f32_16x16x4_f32(
        false, a, false, b, (short)0, acc, false, false);
  }
  float bcol = bias[col0 + lrow];
  int rbase = row0 + ((lane >> 4) << 3);
  #pragma unroll
  for (int r = 0; r < 8; ++r)
    C[(size_t)(rbase + r) * DD + col0 + lrow] = acc[r] + bcol;
}

// ---------- rmsnorm (+optional add, +optional mish), one wave per row ------
__global__ __launch_bounds__(256) void rownorm_kernel(
    const float* __restrict__ X, const float* __restrict__ Yadd,
    const float* __restrict__ scale, float* __restrict__ out,
    int nrows, int domish) {
  int wid = (int)((blockIdx.x * blockDim.x + threadIdx.x) >> 5);
  if (wid >= nrows) return;
  int lane = threadIdx.x & 31;
  float4 v = ((const float4*)(X + (size_t)wid * DD))[lane];
  if (Yadd) {
    float4 w = ((const float4*)(Yadd + (size_t)wid * DD))[lane];
    v.x += w.x; v.y += w.y; v.z += w.z; v.w += w.w;
  }
  float ss = v.x*v.x + v.y*v.y + v.z*v.z + v.w*v.w;
  ss = wave_red_add(ss);
  float r = rsqrtf(ss * (1.0f / DD) + 1e-6f);
  float4 s4 = ((const float4*)scale)[lane];
  v.x *= r * s4.x; v.y *= r * s4.y; v.z *= r * s4.z; v.w *= r * s4.w;
  if (domish) { v.x = mish_f(v.x); v.y = mish_f(v.y); v.z = mish_f(v.z); v.w = mish_f(v.w); }
  ((float4*)(out + (size_t)wid * DD))[lane] = v;
}

// ---------- edge pass 1: logits + segment max (1 wave / edge) --------------
__global__ __launch_bounds__(256) void edge_logits_kernel(
    const float* __restrict__ efeat, const unsigned char* __restrict__ emask,
    const int* __restrict__ elist,
    const float* __restrict__ qb, const float* __restrict__ kb,
    const float* __restrict__ We, const float* __restrict__ be,
    const float* __restrict__ attn_a,
    float* __restrict__ logits, unsigned* __restrict__ lmax) {
  int gw = (int)((blockIdx.x * blockDim.x + threadIdx.x) >> 5);
  if (gw >= BB * EE) return;
  int lane = threadIdx.x & 31;
  int b = gw >> 15;
  bool m = emask[gw] != 0;
  int s = m ? elist[2 * gw]     : (NP1 - 1);
  int r = m ? elist[2 * gw + 1] : (NP1 - 1);
  size_t srow = (size_t)b * NP1 + s;
  size_t rrow = (size_t)b * NP1 + r;
  int d0 = lane * 4;
  int head = lane >> 2;
  float4 q4 = *(const float4*)(qb + rrow * DD + d0);
  float4 k4 = *(const float4*)(kb + srow * DD + d0);
  float xin[8];
  const float* ep = efeat + (size_t)gw * 8;
  #pragma unroll
  for (int i = 0; i < 8; ++i) xin[i] = ep[i];
  float f[4];
  #pragma unroll
  for (int c = 0; c < 4; ++c) {
    int d = d0 + c;
    float acc = be[d];
    #pragma unroll
    for (int i = 0; i < 8; ++i) acc += xin[i] * We[i * DD + d];
    f[c] = acc;
  }
  f[0] += q4.x + k4.x; f[1] += q4.y + k4.y;
  f[2] += q4.z + k4.z; f[3] += q4.w + k4.w;
  float logit = 0.0f;
  #pragma unroll
  for (int c = 0; c < 4; ++c) {
    float v = f[c];
    v = (v > 0.0f) ? v : 0.2f * v;                // leaky_relu 0.2
    logit += v * attn_a[head * DH + (d0 & 15) + c];
  }
  logit += __shfl_xor(logit, 1, 32);
  logit += __shfl_xor(logit, 2, 32);
  if ((lane & 3) == 0) {
    logits[(size_t)gw * HH + head] = logit;
    atomicMax(lmax + rrow * HH + head, f2ord(logit));
  }
}

// ---------- edge pass 2: exp + segment sum (1 thread / edge-head) ----------
__global__ __launch_bounds__(256) void edge_exp_kernel(
    const unsigned char* __restrict__ emask, const int* __restrict__ elist,
    const unsigned* __restrict__ lmax, float* __restrict__ logits,
    float* __restrict__ den) {
  size_t idx = (size_t)blockIdx.x * blockDim.x + threadIdx.x;
  if (idx >= (size_t)BB * EE * HH) return;
  size_t e = idx >> 3;
  int head = (int)(idx & 7);
  int b = (int)(e >> 15);
  bool m = emask[e] != 0;
  int r = m ? elist[2 * e + 1] : (NP1 - 1);
  size_t rrow = (size_t)b * NP1 + r;
  float lm = ord2f(lmax[rrow * HH + head]);
  if (!isfinite(lm)) lm = 0.0f;                   // reference: where(isfinite)
  float ex = __expf(logits[idx] - lm);
  logits[idx] = ex;                               // store ex in place
  atomicAdd(den + rrow * HH + head, ex);
}

// ---------- edge pass 3: alpha * k scatter (1 wave / edge) -----------------
__global__ __launch_bounds__(256) void edge_scatter_kernel(
    const unsigned char* __restrict__ emask, const int* __restrict__ elist,
    const float* __restrict__ exb, const float* __restrict__ den,
    const float* __restrict__ kb, float* __restrict__ hg) {
  int gw = (int)((blockIdx.x * blockDim.x + threadIdx.x) >> 5);
  if (gw >= BB * EE) return;
  int lane = threadIdx.x & 31;
  int b = gw >> 15;
  bool m = emask[gw] != 0;
  int s = m ? elist[2 * gw]     : (NP1 - 1);
  int r = m ? elist[2 * gw + 1] : (NP1 - 1);
  size_t srow = (size_t)b * NP1 + s;
  size_t rrow = (size_t)b * NP1 + r;
  int d0 = lane * 4;
  int head = lane >> 2;
  float ex = exb[(size_t)gw * HH + head];
  float dn = den[rrow * HH + head];
  float alpha = ex / (dn + 1e-9f);
  float4 k4 = *(const float4*)(kb + srow * DD + d0);
  float* dst = hg + rrow * DD + d0;
  atomicAdd(dst + 0, alpha * k4.x);
  atomicAdd(dst + 1, alpha * k4.y);
  atomicAdd(dst + 2, alpha * k4.z);
  atomicAdd(dst + 3, alpha * k4.w);
}

// ---------- pq = seed @ pma_q + b (one tiny block) -------------------------
__global__ void pq_kernel(const float* __restrict__ seed,
                          const float* __restrict__ Wq,
                          const float* __restrict__ bq,
                          float* __restrict__ pq) {
  int j = threadIdx.x;
  float acc = bq[j];
  for (int i = 0; i < DD; ++i) acc += seed[i] * Wq[i * DD + j];
  pq[j] = acc;
}

// ---------- PMA pooling: one block per (batch, head) -----------------------
__global__ __launch_bounds__(256) void pma_pool_kernel(
    const float* __restrict__ pq, const float* __restrict__ pk,
    const float* __restrict__ pv, const unsigned char* __restrict__ nmask,
    float* __restrict__ pooled) {
  __shared__ float sc[NP1];
  __shared__ float red[256];
  __shared__ float acc_s[DH];
  int b = blockIdx.x >> 3;
  int head = blockIdx.x & 7;
  int t = threadIdx.x;
  float pqv[DH];
  #pragma unroll
  for (int d = 0; d < DH; ++d) pqv[d] = pq[head * DH + d];

  for (int n = t; n < NP1; n += 256) {
    bool valid = (n < NN) && (nmask[(size_t)b * NN + n] != 0);
    float s = -1e30f;
    if (valid) {
      const float* kp = pk + ((size_t)b * NP1 + n) * DD + head * DH;
      float a = 0.0f;
      #pragma unroll
      for (int d = 0; d < DH; ++d) a += pqv[d] * kp[d];
      s = a * 0.25f;                   // 1/sqrt(DH)
    }
    sc[n] = s;
  }
  __syncthreads();
  float mx = -1e30f;
  for (int n = t; n < NP1; n += 256) mx = fmaxf(mx, sc[n]);
  red[t] = mx; __syncthreads();
  for (int s2 = 128; s2 > 0; s2 >>= 1) {
    if (t < s2) red[t] = fmaxf(red[t], red[t + s2]);
    __syncthreads();
  }
  mx = red[0]; __syncthreads();
  float sum = 0.0f;
  for (int n = t; n < NP1; n += 256) {
    float e = __expf(sc[n] - mx);
    sc[n] = e;
    sum += e;
  }
  red[t] = sum; __syncthreads();
  for (int s2 = 128; s2 > 0; s2 >>= 1) {
    if (t < s2) red[t] += red[t + s2];
    __syncthreads();
  }
  float inv = 1.0f / red[0];
  if (t < DH) acc_s[t] = 0.0f;
  __syncthreads();
  float part[DH];
  #pragma unroll
  for (int d = 0; d < DH; ++d) part[d] = 0.0f;
  for (int n = t; n < NP1; n += 256) {
    float a = sc[n] * inv;
    const float* vp = pv + ((size_t)b * NP1 + n) * DD + head * DH;
    #pragma unroll
    for (int d = 0; d < DH; ++d) part[d] += a * vp[d];
  }
  #pragma unroll
  for (int d = 0; d < DH; ++d) atomicAdd(&acc_s[d], part[d]);
  __syncthreads();
  if (t < DH) pooled[(size_t)b * DD + head * DH + t] = acc_s[t];
}

// ---------- final: out = [pooled @ pma_o + b, global_features] -------------
__global__ void final_out_kernel(const float* __restrict__ pooled,
                                 const float* __restrict__ Wo,
                                 const float* __restrict__ bo,
                                 const float* __restrict__ gfeat,
                                 float* __restrict__ out) {
  int b = blockIdx.x;
  int t = threadIdx.x;
  if (t < DD) {
    float acc = bo[t];
    const float* pp = pooled + (size_t)b * DD;
    for (int i = 0; i < DD; ++i) acc += pp[i] * Wo[i * DD + t];
    out[(size_t)b * 144 + t] = acc;
  } else if (t < 144) {
    out[(size_t)b * 144 + t] = gfeat[(size_t)b * 16 + (t - DD)];
  }
}

// ===========================================================================
extern "C" void kernel_launch(void* const* d_in, const int* in_sizes, int n_in,
                              void* d_out, int out_size, void* d_ws, size_t ws_size,
                              hipStream_t stream) {
  (void)in_sizes; (void)n_in; (void)out_size; (void)ws_size;
  // ---- inputs (setup_inputs dict order; params pytree in sorted-key order)
  const float*         nf    = (const float*)d_in[0];
  const unsigned char* nmask = (const unsigned char*)d_in[1];
  const float*         efeat = (const float*)d_in[2];
  const unsigned char* emask = (const unsigned char*)d_in[3];
  const int*           elist = (const int*)d_in[4];
  const float*         gfeat = (const float*)d_in[5];
  const float* e1b = (const float*)d_in[6];
  const float* e1w = (const float*)d_in[7];
  const float* e2b = (const float*)d_in[8];
  const float* e2w = (const float*)d_in[9];
  // per layer (sorted keys): gat_a, gat_e{b,w}, gat_k{b,w}, gat_q{b,w}, norm,
  //                          pma_k{b,w}, pma_o{b,w}, pma_q{b,w}, pma_seed,
  //                          pma_v{b,w}, skip{b,w}  -> 19 tensors
  const float* norm1 = (const float*)d_in[10 + 19 * 3];
  const float* norm2 = (const float*)d_in[10 + 19 * 3 + 1];

  // ---- workspace carve-up (floats)
  float* ws = (float*)d_ws;
  float*    h    = ws;                       // MP*128
  float*    skp  = h    + (size_t)MP * DD;
  float*    qb   = skp  + (size_t)MP * DD;
  float*    kb   = qb   + (size_t)MP * DD;
  float*    hg   = kb   + (size_t)MP * DD;
  float*    exb  = hg   + (size_t)MP * DD;   // B*E*H
  unsigned* lmax = (unsigned*)(exb + (size_t)BB * EE * HH);  // MP*H
  float*    den  = (float*)lmax + (size_t)MP * HH;
  float*    pq   = den + (size_t)MP * HH;    // 128
  float*    pooled = pq + DD;                // B*128
  float*    outp = (float*)d_out;

  const int TB = 256;
  const int gRowWave  = (ROWS * 32 + TB - 1) / TB;   // 1 wave / valid row
  const int gRowWaveP = (MP * 32 + TB - 1) / TB;     // 1 wave / padded row
  const int gGemm     = MP / 16;                     // 1025 blocks
  const int gEdgeWave = (BB * EE * 32) / TB;         // 32768 blocks
  const int gEdgeHead = (BB * EE * HH) / TB;         // 8192 blocks

  // ---- embedding: tmp(hg) = mish(rmsnorm(nodes@W1+b1)); h = rmsnorm(tmp@W2+b2)
  fill_f32_kernel<<<4096, TB, 0, stream>>>(hg, 0.0f, (size_t)MP * DD);
  embed1_kernel<<<gRowWave, TB, 0, stream>>>(nf, e1w, e1b, norm1, hg);
  gemm128_kernel<<<gGemm, TB, 0, stream>>>(hg, e2w, e2b, h);
  rownorm_kernel<<<gRowWaveP, TB, 0, stream>>>(h, nullptr, norm2, h, MP, 0);

  for (int l = 0; l < 3; ++l) {
    int base = 10 + 19 * l;
    const float* gat_a  = (const float*)d_in[base + 0];
    const float* gat_eb = (const float*)d_in[base + 1];
    const float* gat_ew = (const float*)d_in[base + 2];
    const float* gat_kb = (const float*)d_in[base + 3];
    const float* gat_kw = (const float*)d_in[base + 4];
    const float* gat_qb = (const float*)d_in[base + 5];
    const float* gat_qw = (const float*)d_in[base + 6];
    const float* lnorm  = (const float*)d_in[base + 7];
    const float* pmakb  = (const float*)d_in[base + 8];
    const float* pmakw  = (const float*)d_in[base + 9];
    const float* pmaob  = (const float*)d_in[base + 10];
    const float* pmaow  = (const float*)d_in[base + 11];
    const float* pmaqb  = (const float*)d_in[base + 12];
    const float* pmaqw  = (const float*)d_in[base + 13];
    const float* pseed  = (const float*)d_in[base + 14];
    const float* pmavb  = (const float*)d_in[base + 15];
    const float* pmavw  = (const float*)d_in[base + 16];
    const float* skipb  = (const float*)d_in[base + 17];
    const float* skipw  = (const float*)d_in[base + 18];

    // dense projections (WMMA)
    gemm128_kernel<<<gGemm, TB, 0, stream>>>(h, skipw, skipb, skp);
    gemm128_kernel<<<gGemm, TB, 0, stream>>>(h, gat_qw, gat_qb, qb);
    gemm128_kernel<<<gGemm, TB, 0, stream>>>(h, gat_kw, gat_kb, kb);

    // segment buffers
    fill_u32_kernel<<<512,  TB, 0, stream>>>(lmax, ORD_NEG_INF, (size_t)MP * HH);
    fill_f32_kernel<<<512,  TB, 0, stream>>>(den, 0.0f, (size_t)MP * HH);
    fill_f32_kernel<<<4096, TB, 0, stream>>>(hg, 0.0f, (size_t)MP * DD);

    // GATv2 edge softmax (3 passes)
    edge_logits_kernel<<<gEdgeWave, TB, 0, stream>>>(
        efeat, emask, elist, qb, kb, gat_ew, gat_eb, gat_a, exb, lmax);
    edge_exp_kernel<<<gEdgeHead, TB, 0, stream>>>(emask, elist, lmax, exb, den);
    edge_scatter_kernel<<<gEdgeWave, TB, 0, stream>>>(emask, elist, exb, den, kb, hg);

    // h = mish(rmsnorm(hg + skip))
    rownorm_kernel<<<gRowWaveP, TB, 0, stream>>>(hg, skp, lnorm, h, MP, 1);

    // PMA pooling only matters on the final layer (g is overwritten each iter)
    if (l == 2) {
      gemm128_kernel<<<gGemm, TB, 0, stream>>>(h, pmakw, pmakb, qb);  // pk
      gemm128_kernel<<<gGemm, TB, 0, stream>>>(h, pmavw, pmavb, kb);  // pv
      pq_kernel<<<1, DD, 0, stream>>>(pseed, pmaqw, pmaqb, pq);
      pma_pool_kernel<<<BB * HH, TB, 0, stream>>>(pq, qb, kb, nmask, pooled);
      final_out_kernel<<<BB, 160, 0, stream>>>(pooled, pmaow, pmaob, gfeat, outp);
    }
  }
}